// CategoryMultiplier_9019431321846
// MI455X (gfx1250) — compile-verified
//
#include <hip/hip_runtime.h>
#include <stdint.h>

// out[b,s,d] = inputs[b,s,d] * (cat!=0 ? emb_table[cat,d] : 1.0f)
// Pure bandwidth problem: ~270 MB HBM traffic, ~11.6 us floor at 23.3 TB/s.
// gfx1250-specific paths used:
//  - global_load_async_to_lds_b128 (ASYNCcnt) to stage the gathered table row
//  - s_wait_asynccnt
//  - non-temporal b128 stream loads/stores so the 2MB table stays L2-resident

typedef float v4f __attribute__((ext_vector_type(4)));

#define BB   128
#define SS   512
#define DD   512
#define NROW (BB * SS)        // 65536 positions
#define D4   (DD / 4)         // 128 float4 per row
#define RPB  2                // rows per block
#define TPB  (RPB * D4)       // 256 threads = 8 wave32

__global__ __launch_bounds__(TPB)
void CategoryMultiplier_kernel(const float* __restrict__ inputs,
                               const int*   __restrict__ categories,
                               const float* __restrict__ emb_table,
                               float*       __restrict__ out)
{
    __shared__ float lds_row[RPB][DD];   // 4 KB staging for gathered emb rows

    const int tid  = threadIdx.x;
    const int rloc = tid >> 7;           // 0..1: which row in this block
    const int c4   = tid & (D4 - 1);     // float4 column index within row
    const int row  = blockIdx.x * RPB + rloc;

    const int cat = categories[row];     // 0 == padding (row 0 is still a valid gather)

    // --- CDNA5 async copy: 16B slice of emb_table[cat] -> LDS (ASYNCcnt) ---
    const float* gsrc   = emb_table + (size_t)cat * DD + (c4 << 2);
    uint32_t lds_off    = (uint32_t)(uintptr_t)(&lds_row[rloc][c4 << 2]);
    uint64_t gaddr      = (uint64_t)(uintptr_t)gsrc;
    asm volatile("global_load_async_to_lds_b128 %0, %1, off"
                 :
                 : "v"(lds_off), "v"(gaddr)
                 : "memory");

    // --- Stream the input slice non-temporally (read-once, don't pollute L2) ---
    const size_t idx4 = (size_t)row * D4 + c4;
    v4f x = __builtin_nontemporal_load((const v4f*)inputs + idx4);

    // Wait for our own async transfer; each lane reads only the LDS bytes
    // it copied, so no s_barrier is required.
    asm volatile("s_wait_asynccnt 0" ::: "memory");

    v4f m = *(const v4f*)&lds_row[rloc][c4 << 2];
    const v4f one = {1.0f, 1.0f, 1.0f, 1.0f};
    m = (cat == 0) ? one : m;            // branch-free v_cndmask select

    v4f r = x * m;
    __builtin_nontemporal_store(r, (v4f*)out + idx4);
}

extern "C" void kernel_launch(void* const* d_in, const int* in_sizes, int n_in,
                              void* d_out, int out_size, void* d_ws, size_t ws_size,
                              hipStream_t stream)
{
    const float* inputs     = (const float*)d_in[0];   // [B,S,D] f32
    const int*   categories = (const int*)  d_in[1];   // [B,S] int
    // d_in[2] = mask_positions: unused by the reference output
    const float* emb_table  = (const float*)d_in[3];   // [VOCAB,D] f32
    float*       out        = (float*)d_out;           // [B,S,D] f32

    dim3 grid(NROW / RPB);   // 32768 blocks
    dim3 block(TPB);         // 256 threads
    CategoryMultiplier_kernel<<<grid, block, 0, stream>>>(inputs, categories, emb_table, out);
}